// GraphConvNet_2456721293646
// MI455X (gfx1250) — compile-verified
//
#include <hip/hip_runtime.h>
#include <math.h>

typedef __attribute__((ext_vector_type(2))) float v2f;
typedef __attribute__((ext_vector_type(8))) float v8f;

#define N_NODES 100000
#define N_EDGES 3200000
#define HID     64
#define NTILES  (N_NODES / 16)   // 6250, exact

// ---------------------------------------------------------------- utilities
__global__ void zero_kernel(float* __restrict__ p, int n) {
    int i = blockIdx.x * blockDim.x + threadIdx.x;
    if (i < n) p[i] = 0.0f;
}

__global__ void degree_kernel(const int* __restrict__ src, const int* __restrict__ dst,
                              float* __restrict__ degOut, float* __restrict__ degIn) {
    int e = blockIdx.x * blockDim.x + threadIdx.x;
    if (e < N_EDGES) {
        atomicAdd(&degOut[src[e]], 1.0f);
        atomicAdd(&degIn[dst[e]], 1.0f);
    }
}

__global__ void norm_kernel(float* __restrict__ a, float* __restrict__ b) {
    int i = blockIdx.x * blockDim.x + threadIdx.x;
    if (i < N_NODES) {
        a[i] = rsqrtf(fmaxf(a[i], 1.0f));
        b[i] = rsqrtf(fmaxf(b[i], 1.0f));
    }
}

// ------------------------------------------------- edge phase (message pass)
// F=3 input features: one thread per edge, 3 atomics.
__global__ void scatter3_kernel(const int* __restrict__ src, const int* __restrict__ dst,
                                const float* __restrict__ feat,
                                const float* __restrict__ normSrc,
                                float* __restrict__ agg) {
    int e = blockIdx.x * blockDim.x + threadIdx.x;
    if (e >= N_EDGES) return;
    int s = src[e], d = dst[e];
    float ns = normSrc[s];
    atomicAdd(&agg[d * 3 + 0], feat[s * 3 + 0] * ns);
    atomicAdd(&agg[d * 3 + 1], feat[s * 3 + 1] * ns);
    atomicAdd(&agg[d * 3 + 2], feat[s * 3 + 2] * ns);
}

// F=64: 16 threads per edge, float4 gather (coalesced 256B row), 4 float atomics.
__global__ void scatter64_kernel(const int* __restrict__ src, const int* __restrict__ dst,
                                 const float* __restrict__ x,
                                 const float* __restrict__ normSrc,
                                 float* __restrict__ agg) {
    long long t = (long long)blockIdx.x * blockDim.x + threadIdx.x;
    if (t >= (long long)N_EDGES * 16) return;
    int e = (int)(t >> 4);
    int q = (int)(t & 15);
    int s = src[e], d = dst[e];
    float ns = normSrc[s];
    const float4 v = *(const float4*)(x + (size_t)s * HID + q * 4);
    float* o = agg + (size_t)d * HID + q * 4;
    atomicAdd(o + 0, v.x * ns);
    atomicAdd(o + 1, v.y * ns);
    atomicAdd(o + 2, v.z * ns);
    atomicAdd(o + 3, v.w * ns);
}

// ------------------------------------------------------------- layer 0 (K=3)
__global__ void dense0_kernel(const float* __restrict__ agg3,
                              const float* __restrict__ normDst,
                              const float* __restrict__ W0,   // [3][64]
                              const float* __restrict__ b0,   // [64]
                              float* __restrict__ x) {
    int t = blockIdx.x * blockDim.x + threadIdx.x;
    if (t >= N_NODES * HID) return;
    int m = t >> 6, n = t & 63;
    float nd = normDst[m];
    float a0 = agg3[m * 3 + 0], a1 = agg3[m * 3 + 1], a2 = agg3[m * 3 + 2];
    float v = a0 * W0[n] + a1 * W0[HID + n] + a2 * W0[2 * HID + n];
    v = v * nd + b0[n];
    x[t] = fmaxf(v, 0.0f);
}

// --------------------------------------------- hidden GEMM: [N,64]@[64,64]
// One wave32 computes a 16-row tile. fp32 WMMA 16x16x4, 16 K-steps, 4 col tiles.
// A layout (32-bit A 16x4): lane&15 = M row; lanes 0-15 hold K {0,1}, lanes
// 16-31 hold K {2,3} of each K-step. D layout: VGPR r -> row r (lanes 0-15)
// or 8+r (lanes 16-31), col = lane&15.
__global__ void gemm_hidden_kernel(const float* __restrict__ agg,
                                   const float* __restrict__ normDst,
                                   const float* __restrict__ W,    // [64][64]
                                   const float* __restrict__ bias, // [64]
                                   float* __restrict__ xout) {
    __shared__ float Wlds[HID * HID];
    for (int i = threadIdx.x; i < HID * HID / 4; i += blockDim.x)
        ((float4*)Wlds)[i] = ((const float4*)W)[i];
    __syncthreads();

    const int wave = threadIdx.x >> 5;
    const int lane = threadIdx.x & 31;
    const int tile = blockIdx.x * (blockDim.x >> 5) + wave;
    if (tile >= NTILES) return;   // wave-uniform: EXEC all-1 for WMMA below

    const int rowA = lane & 15;
    const int koff = (lane >> 4) * 2;          // 0 or 2
    const int m    = tile * 16 + rowA;
    const float nd = normDst[m];
    const float* arow = agg + (size_t)m * HID;

    v2f a[16];
#pragma unroll
    for (int ks = 0; ks < 16; ++ks) {
        float2 t2 = *(const float2*)(arow + ks * 4 + koff);
        a[ks].x = t2.x * nd;
        a[ks].y = t2.y * nd;
    }

    const int n0     = lane & 15;
    const int mstore = tile * 16 + (lane >> 4) * 8;
#pragma unroll
    for (int ct = 0; ct < 4; ++ct) {
        const int n = ct * 16 + n0;
        v8f acc = {};
#pragma unroll
        for (int ks = 0; ks < 16; ++ks) {
            const int kk = ks * 4 + koff;
            v2f b;
            b.x = Wlds[kk * HID + n];
            b.y = Wlds[(kk + 1) * HID + n];
            acc = __builtin_amdgcn_wmma_f32_16x16x4_f32(
                false, a[ks], false, b, (short)0, acc, false, false);
        }
        const float bn = bias[n];
#pragma unroll
        for (int r = 0; r < 8; ++r) {
            float v = acc[r] + bn;
            xout[(size_t)(mstore + r) * HID + n] = fmaxf(v, 0.0f);
        }
    }
}

// --------------------------------------- output GEMM: [N,64]@[64,6]+sigmoid
__global__ void gemm_out_kernel(const float* __restrict__ agg,
                                const float* __restrict__ normDst,
                                const float* __restrict__ Wout, // [64][6]
                                const float* __restrict__ bout, // [6]
                                float* __restrict__ out) {      // [N][6]
    __shared__ float Wlds[HID * 16];  // padded to 16 cols with zeros
    for (int i = threadIdx.x; i < HID * 16; i += blockDim.x) {
        int k = i >> 4, n = i & 15;
        Wlds[i] = (n < 6) ? Wout[k * 6 + n] : 0.0f;
    }
    __syncthreads();

    const int wave = threadIdx.x >> 5;
    const int lane = threadIdx.x & 31;
    const int tile = blockIdx.x * (blockDim.x >> 5) + wave;
    if (tile >= NTILES) return;

    const int rowA = lane & 15;
    const int koff = (lane >> 4) * 2;
    const int m    = tile * 16 + rowA;
    const float nd = normDst[m];
    const float* arow = agg + (size_t)m * HID;
    const int n0 = lane & 15;

    v8f acc = {};
#pragma unroll
    for (int ks = 0; ks < 16; ++ks) {
        const int kk = ks * 4 + koff;
        float2 t2 = *(const float2*)(arow + ks * 4 + koff);
        v2f a; a.x = t2.x * nd; a.y = t2.y * nd;
        v2f b;
        b.x = Wlds[kk * 16 + n0];
        b.y = Wlds[(kk + 1) * 16 + n0];
        acc = __builtin_amdgcn_wmma_f32_16x16x4_f32(
            false, a, false, b, (short)0, acc, false, false);
    }
    if (n0 < 6) {
        const int mstore = tile * 16 + (lane >> 4) * 8;
        const float bn = bout[n0];
#pragma unroll
        for (int r = 0; r < 8; ++r) {
            float v = acc[r] + bn;
            v = 1.0f / (1.0f + __expf(-v));
            out[(size_t)(mstore + r) * 6 + n0] = v;
        }
    }
}

// ------------------------------------------------------------------- driver
extern "C" void kernel_launch(void* const* d_in, const int* in_sizes, int n_in,
                              void* d_out, int out_size, void* d_ws, size_t ws_size,
                              hipStream_t stream) {
    (void)in_sizes; (void)n_in; (void)out_size; (void)ws_size;

    const float* feat = (const float*)d_in[0];
    const float* W0   = (const float*)d_in[1];
    const float* b0   = (const float*)d_in[2];
    const float* Wh   = (const float*)d_in[3];
    const float* bh   = (const float*)d_in[4];
    const float* Wout = (const float*)d_in[5];
    const float* bout = (const float*)d_in[6];
    const int*   src  = (const int*)d_in[7];
    const int*   dst  = (const int*)d_in[8];
    float* out = (float*)d_out;

    float* ws      = (float*)d_ws;
    float* normSrc = ws;                                 // N
    float* normDst = ws + N_NODES;                       // N
    float* agg     = ws + 2 * (size_t)N_NODES;           // 64N
    float* x       = agg + (size_t)N_NODES * HID;        // 64N

    const int B = 256;
    const int zeroBig   = (N_NODES * HID + B - 1) / B;
    const int edgeBlks  = (N_EDGES + B - 1) / B;
    const long long st  = (long long)N_EDGES * 16;
    const int scatBlks  = (int)((st + B - 1) / B);
    const int gemmBlks  = (NTILES + 7) / 8;              // 8 waves per block

    // degrees -> norms
    zero_kernel<<<(2 * N_NODES + B - 1) / B, B, 0, stream>>>(normSrc, 2 * N_NODES);
    degree_kernel<<<edgeBlks, B, 0, stream>>>(src, dst, normSrc, normDst);
    norm_kernel<<<(N_NODES + B - 1) / B, B, 0, stream>>>(normSrc, normDst);

    // layer 0 (features F=3)
    zero_kernel<<<(3 * N_NODES + B - 1) / B, B, 0, stream>>>(agg, 3 * N_NODES);
    scatter3_kernel<<<edgeBlks, B, 0, stream>>>(src, dst, feat, normSrc, agg);
    dense0_kernel<<<(N_NODES * HID + B - 1) / B, B, 0, stream>>>(agg, normDst, W0, b0, x);

    // 6 hidden layers
    for (int i = 0; i < 6; ++i) {
        zero_kernel<<<zeroBig, B, 0, stream>>>(agg, N_NODES * HID);
        scatter64_kernel<<<scatBlks, B, 0, stream>>>(src, dst, x, normSrc, agg);
        gemm_hidden_kernel<<<gemmBlks, B, 0, stream>>>(
            agg, normDst, Wh + (size_t)i * HID * HID, bh + (size_t)i * HID, x);
    }

    // output layer
    zero_kernel<<<zeroBig, B, 0, stream>>>(agg, N_NODES * HID);
    scatter64_kernel<<<scatBlks, B, 0, stream>>>(src, dst, x, normSrc, agg);
    gemm_out_kernel<<<gemmBlks, B, 0, stream>>>(agg, normDst, Wout, bout, out);
}